// GATExtractPart_18176301596828
// MI455X (gfx1250) — compile-verified
//
#include <hip/hip_runtime.h>
#include <hip/hip_bf16.h>
#include <float.h>

// ---------------------------------------------------------------------------
// CDNA5 (gfx1250) GAT forward. Wave32. Matrix ops via V_WMMA_F32_16X16X4_F32.
// ---------------------------------------------------------------------------

typedef __attribute__((ext_vector_type(2))) float v2f;
typedef __attribute__((ext_vector_type(8))) float v8f;

#define WARP 32

__device__ __forceinline__ float wred32(float v) {
    for (int m = 16; m > 0; m >>= 1) v += __shfl_xor(v, m, 32);
    return v;
}

__device__ __forceinline__ void atomicMaxF(float* addr, float v) {
    if (v >= 0.0f) atomicMax((int*)addr, __float_as_int(v));
    else           atomicMin((unsigned int*)addr, __float_as_uint(v));
}

__device__ __forceinline__ void edge_sd(int ef, const int* ei, int E, int& s, int& d) {
    if (ef < E) { s = ei[ef]; d = ei[E + ef]; }
    else        { s = ef - E; d = ef - E; }
}

// ---------------------------------------------------------------------------
// small utility kernels
// ---------------------------------------------------------------------------
__global__ void k_fill_f(float* p, float v, int n) {
    int i = blockIdx.x * blockDim.x + threadIdx.x;
    if (i < n) p[i] = v;
}
__global__ void k_fill_i(int* p, int v, int n) {
    int i = blockIdx.x * blockDim.x + threadIdx.x;
    if (i < n) p[i] = v;
}

// degree + edge-attr sum per destination (original edges only)
__global__ void k_deg_ea(const int* ei, const float* ea, int* degI, float* easum, int E) {
    int e = blockIdx.x * blockDim.x + threadIdx.x;
    if (e >= E) return;
    int d = ei[E + e];
    atomicAdd(&degI[d], 1);
#pragma unroll
    for (int j = 0; j < 6; j++) atomicAdd(&easum[d * 6 + j], ea[(size_t)e * 6 + j]);
}

// mean edge attr (self-loop fill_value='mean'); in-place on easum
__global__ void k_mean_ea(float* easum, const int* degI, int N) {
    int i = blockIdx.x * blockDim.x + threadIdx.x;
    if (i >= N * 6) return;
    float dg = (float)degI[i / 6];
    easum[i] = easum[i] / fmaxf(dg, 1.0f);
}

// single-block exclusive scan of (deg[n]+1) -> CSR offsets (N = 32768, 1024 thr x 32)
__global__ void k_scan(const int* degI, int* offsets, int N) {
    __shared__ int part[1024];
    int t = threadIdx.x;
    int chunk = N / 1024;
    int base = t * chunk;
    int s = 0;
    for (int i = 0; i < chunk; i++) s += degI[base + i] + 1;
    part[t] = s; __syncthreads();
    for (int d = 1; d < 1024; d <<= 1) {
        int v = (t >= d) ? part[t - d] : 0;
        __syncthreads();
        part[t] += v;
        __syncthreads();
    }
    int run = (t == 0) ? 0 : part[t - 1];
    for (int i = 0; i < chunk; i++) { offsets[base + i] = run; run += degI[base + i] + 1; }
    if (t == 1023) offsets[N] = run;
}

__global__ void k_csr_fill(const int* ei, const int* offsets, int* cursor,
                           int* csr_e, int* csr_s, int E, int EF) {
    int ef = blockIdx.x * blockDim.x + threadIdx.x;
    if (ef >= EF) return;
    int s, d; edge_sd(ef, ei, E, s, d);
    int pos = atomicAdd(&cursor[d], 1);
    int idx = offsets[d] + pos;
    csr_e[idx] = ef;
    csr_s[idx] = s;
}

// we_ae[k*H+h] = sum_c We[k, h*256+c] * ae[h*256+c]   (block per output)
__global__ void k_weae(const float* We, const float* ae, float* out, int H, int ldWe) {
    int o = blockIdx.x;
    int k = o / H, hh = o % H;
    int t = threadIdx.x;
    __shared__ float sb[256];
    sb[t] = We[(size_t)k * ldWe + hh * 256 + t] * ae[hh * 256 + t];
    __syncthreads();
    for (int st = 128; st; st >>= 1) { if (t < st) sb[t] += sb[t + st]; __syncthreads(); }
    if (t == 0) out[o] = sb[0];
}

// per-row dot of [NN, H*256] with [H,256] vector -> out[n*H+h]  (warp per (n,h))
__global__ void k_rowdot(const float* h, const float* avec, float* out, int NN, int H) {
    int wid  = (blockIdx.x * blockDim.x + threadIdx.x) >> 5;
    int lane = threadIdx.x & 31;
    if (wid >= NN * H) return;
    int n = wid / H, hh = wid - n * H;
    const float* row = h + (size_t)n * H * 256 + hh * 256;
    const float* a   = avec + hh * 256;
    float s = 0.0f;
#pragma unroll
    for (int i = 0; i < 8; i++) s += row[i * 32 + lane] * a[i * 32 + lane];
    s = wred32(s);
    if (lane == 0) out[wid] = s;
}

// edge attention logits + segment max
__global__ void k_logits(const int* ei, const float* ea, const float* eamean,
                         const float* hs, const float* hd, const float* weae,
                         float* att, float* segmax, int E, int N, int H) {
    int idx = blockIdx.x * blockDim.x + threadIdx.x;
    int total = (E + N) * H;
    if (idx >= total) return;
    int ef = idx / H, hh = idx - ef * H;
    int s, d; edge_sd(ef, ei, E, s, d);
    const float* eav = (ef < E) ? (ea + (size_t)ef * 6) : (eamean + (size_t)(ef - E) * 6);
    float al = hs[s * H + hh] + hd[d * H + hh];
#pragma unroll
    for (int k = 0; k < 6; k++) al += eav[k] * weae[k * H + hh];
    al = (al > 0.0f) ? al : 0.2f * al;            // leaky_relu(0.2)
    att[idx] = al;
    atomicMaxF(&segmax[d * H + hh], al);
}

// exp(al - max) in place + segment sum
__global__ void k_expsum(const int* ei, float* att, const float* segmax,
                         float* segsum, int E, int N, int H) {
    int idx = blockIdx.x * blockDim.x + threadIdx.x;
    int total = (E + N) * H;
    if (idx >= total) return;
    int ef = idx / H, hh = idx - ef * H;
    int s, d; edge_sd(ef, ei, E, s, d);
    float ex = expf(att[idx] - segmax[d * H + hh]);
    att[idx] = ex;
    atomicAdd(&segsum[d * H + hh], ex);
}

// CSR gather aggregation: out[n, h*256+c] = sum_e alpha_norm * hsrc[src, h*256+c]
__global__ void k_aggregate(const int* offsets, const int* csr_e, const int* csr_s,
                            const float* att, const float* segsum, const float* hsrc,
                            float* outp, int H) {
    int n = blockIdx.x;
    int t = threadIdx.x;         // channel within head
    int D = H * 256;
    int beg = offsets[n], end = offsets[n + 1];
    float invs[4], acc[4];
#pragma unroll
    for (int h = 0; h < 4; h++) { acc[h] = 0.0f; invs[h] = 0.0f; }
    for (int h = 0; h < H; h++) invs[h] = 1.0f / (segsum[n * H + h] + 1e-16f);
    for (int p = beg; p < end; p++) {
        int ef = csr_e[p];
        int s  = csr_s[p];
        const float* hrow = hsrc + (size_t)s * D;
        for (int h = 0; h < H; h++) {
            float w = att[ef * H + h] * invs[h];
            acc[h] += w * hrow[h * 256 + t];
        }
    }
    for (int h = 0; h < H; h++) outp[(size_t)n * D + h * 256 + t] = acc[h];
}

// bias + layernorm (+ optional relu), block per node, D = 256*per (per<=4)
__global__ void k_bias_ln(const float* x, const float* bias, const float* g,
                          const float* be, float* out, int D, int relu) {
    int n = blockIdx.x, t = threadIdx.x;
    int per = D >> 8;
    float vals[4];
    float s = 0.0f, s2 = 0.0f;
    for (int i = 0; i < per; i++) {
        float v = x[(size_t)n * D + i * 256 + t] + bias[i * 256 + t];
        vals[i] = v; s += v; s2 += v * v;
    }
    __shared__ float sb[256], sb2[256];
    sb[t] = s; sb2[t] = s2; __syncthreads();
    for (int st = 128; st; st >>= 1) {
        if (t < st) { sb[t] += sb[t + st]; sb2[t] += sb2[t + st]; }
        __syncthreads();
    }
    float mu  = sb[0] / (float)D;
    float var = sb2[0] / (float)D - mu * mu;
    float inv = rsqrtf(var + 1e-5f);
    for (int i = 0; i < per; i++) {
        float v = (vals[i] - mu) * inv * g[i * 256 + t] + be[i * 256 + t];
        if (relu) v = fmaxf(v, 0.0f);
        out[(size_t)n * D + i * 256 + t] = v;
    }
}

// ---------------------------------------------------------------------------
// WMMA f32 GEMM: C[M,Nc] = act( A[M,K] @ W[K,Nc] ), fp32 throughout.
// Block = 256 threads = 8 wave32s; block tile 64x64; wave tile 16x32
// (two 16x16 accumulators); K staged 32 at a time in LDS; inner step 4 via
// V_WMMA_F32_16X16X4_F32.
// ---------------------------------------------------------------------------
#define AS_LD 33   // padded LDS stride to avoid bank conflicts on A-frag reads

__global__ __launch_bounds__(256)
void k_gemm_wmma_f32(const float* __restrict__ A, const float* __restrict__ W,
                     float* __restrict__ C, int M, int K, int Nc, int act) {
    __shared__ float As[64 * AS_LD];
    __shared__ float Ws[32 * 64];

    const int tid  = threadIdx.x;
    const int wave = tid >> 5;
    const int lane = tid & 31;
    const int wr   = wave & 3;       // wave row tile 0..3   (16 rows each)
    const int wc   = wave >> 2;      // wave col tile 0..1   (32 cols each)
    const int lrow = lane & 15;
    const bool hi  = lane >= 16;

    const int rowblock = blockIdx.x * 64;
    const int colblock = blockIdx.y * 64;

    v8f acc0 = {};
    v8f acc1 = {};

    for (int k0 = 0; k0 < K; k0 += 32) {
        // stage A tile 64x32 (zero-pad past K)
        for (int i = tid; i < 64 * 32; i += 256) {
            int r = i >> 5, kk = i & 31;
            int gk = k0 + kk;
            As[r * AS_LD + kk] = (gk < K) ? A[(size_t)(rowblock + r) * K + gk] : 0.0f;
        }
        // stage W tile 32x64
        for (int i = tid; i < 32 * 64; i += 256) {
            int kk = i >> 6, c = i & 63;
            int gk = k0 + kk;
            Ws[kk * 64 + c] = (gk < K) ? W[(size_t)gk * Nc + colblock + c] : 0.0f;
        }
        // hint: prefetch next A tile through gfx1250 global_prefetch path
        if (k0 + 32 < K)
            __builtin_prefetch(&A[(size_t)(rowblock + (tid >> 2)) * K + k0 + 32], 0, 1);
        __syncthreads();

#pragma unroll
        for (int kk = 0; kk < 32; kk += 4) {
            // A fragment: 16x4 f32 -> 2 VGPRs per lane
            v2f a;
            a.x = As[(wr * 16 + lrow) * AS_LD + kk + (hi ? 2 : 0)];
            a.y = As[(wr * 16 + lrow) * AS_LD + kk + (hi ? 3 : 1)];
            // B fragments: 4x16 f32 -> 2 VGPRs per lane, two col tiles
            int cb = wc * 32;
            v2f b0, b1;
            b0.x = Ws[(kk + (hi ? 2 : 0)) * 64 + cb + lrow];
            b0.y = Ws[(kk + (hi ? 3 : 1)) * 64 + cb + lrow];
            b1.x = Ws[(kk + (hi ? 2 : 0)) * 64 + cb + 16 + lrow];
            b1.y = Ws[(kk + (hi ? 3 : 1)) * 64 + cb + 16 + lrow];

            acc0 = __builtin_amdgcn_wmma_f32_16x16x4_f32(
                false, a, false, b0, (short)0, acc0, false, false);
            acc1 = __builtin_amdgcn_wmma_f32_16x16x4_f32(
                false, a, false, b1, (short)0, acc1, false, false);
        }
        __syncthreads();
    }

    // store: acc VGPR r holds rows (wr*16 + r) [lanes 0-15] and (wr*16 + 8 + r) [lanes 16-31]
    const int cb = colblock + wc * 32;
#pragma unroll
    for (int r = 0; r < 8; r++) {
        int row = rowblock + wr * 16 + r + (hi ? 8 : 0);
        float v0 = acc0[r];
        float v1 = acc1[r];
        if (act == 1) { v0 = tanhf(v0); v1 = tanhf(v1); }
        C[(size_t)row * Nc + cb + lrow]      = v0;
        C[(size_t)row * Nc + cb + 16 + lrow] = v1;
    }
}

// ---------------------------------------------------------------------------
// per-graph softmax + attention-weighted max pooling (block per graph, L=64)
// ---------------------------------------------------------------------------
__global__ void k_pool(const float* h2, const float* scores, float* drug, float* attnout) {
    int b = blockIdx.x, t = threadIdx.x;
    __shared__ float sw[64];
    if (t < 64) sw[t] = scores[b * 64 + t];
    __syncthreads();
    if (t == 0) {
        float m = -FLT_MAX;
        for (int l = 0; l < 64; l++) m = fmaxf(m, sw[l]);
        float s = 0.0f;
        for (int l = 0; l < 64; l++) { sw[l] = expf(sw[l] - m); s += sw[l]; }
        float inv = 1.0f / (s + 1e-16f);
        for (int l = 0; l < 64; l++) sw[l] *= inv;
    }
    __syncthreads();
    if (t < 64) attnout[b * 64 + t] = sw[t];
    float m = -FLT_MAX;
    for (int l = 0; l < 64; l++)
        m = fmaxf(m, h2[((size_t)b * 64 + l) * 256 + t] * sw[l]);
    drug[b * 256 + t] = m;
}

__global__ void k_food(const int* food_ids, const float* emb, float* out, int B) {
    int i = blockIdx.x * blockDim.x + threadIdx.x;
    if (i >= B * 256) return;
    out[i] = emb[food_ids[i >> 8] * 256 + (i & 255)];
}

// ---------------------------------------------------------------------------
static inline size_t rnd256(size_t x) { return (x + 255) & ~(size_t)255; }

extern "C" void kernel_launch(void* const* d_in, const int* in_sizes, int n_in,
                              void* d_out, int out_size, void* d_ws, size_t ws_size,
                              hipStream_t stream) {
    const float* x    = (const float*)d_in[0];
    const float* ea   = (const float*)d_in[1];
    const int*   ei   = (const int*)d_in[2];
    const int*   fids = (const int*)d_in[4];
    const float* W1   = (const float*)d_in[5];
    const float* as1  = (const float*)d_in[6];
    const float* ad1  = (const float*)d_in[7];
    const float* We1  = (const float*)d_in[8];
    const float* ae1  = (const float*)d_in[9];
    const float* b1   = (const float*)d_in[10];
    const float* g1   = (const float*)d_in[11];
    const float* be1  = (const float*)d_in[12];
    const float* W2   = (const float*)d_in[13];
    const float* as2  = (const float*)d_in[14];
    const float* ad2  = (const float*)d_in[15];
    const float* We2  = (const float*)d_in[16];
    const float* ae2  = (const float*)d_in[17];
    const float* b2   = (const float*)d_in[18];
    const float* g2   = (const float*)d_in[19];
    const float* be2  = (const float*)d_in[20];
    const float* Wa   = (const float*)d_in[21];
    const float* va   = (const float*)d_in[22];
    const float* emb  = (const float*)d_in[23];

    const int N  = in_sizes[0] / 55;     // 32768
    const int E  = in_sizes[1] / 6;      // 131072
    const int B  = in_sizes[4];          // 512
    const int EF = E + N;                // with self-loops
    const int H1 = 4;

    // workspace carve
    char* w = (char*)d_ws;
    size_t o = 0;
    int*   degI    = (int*)(w + o);   o += rnd256(sizeof(int) * N);
    int*   offsets = (int*)(w + o);   o += rnd256(sizeof(int) * (N + 1));
    int*   cursor  = (int*)(w + o);   o += rnd256(sizeof(int) * N);
    int*   csr_e   = (int*)(w + o);   o += rnd256(sizeof(int) * EF);
    int*   csr_s   = (int*)(w + o);   o += rnd256(sizeof(int) * EF);
    float* eamean  = (float*)(w + o); o += rnd256(sizeof(float) * N * 6);
    float* weae1   = (float*)(w + o); o += rnd256(sizeof(float) * 32);
    float* weae2   = (float*)(w + o); o += rnd256(sizeof(float) * 8);
    float* hs      = (float*)(w + o); o += rnd256(sizeof(float) * N * 4);
    float* hd      = (float*)(w + o); o += rnd256(sizeof(float) * N * 4);
    float* segmax  = (float*)(w + o); o += rnd256(sizeof(float) * N * 4);
    float* segsum  = (float*)(w + o); o += rnd256(sizeof(float) * N * 4);
    float* att     = (float*)(w + o); o += rnd256(sizeof(float) * (size_t)EF * 4);
    float* scores  = (float*)(w + o); o += rnd256(sizeof(float) * N);
    float* h1      = (float*)(w + o); o += rnd256(sizeof(float) * (size_t)N * 1024);
    float* agg1    = (float*)(w + o); o += rnd256(sizeof(float) * (size_t)N * 1024);
    float* h2lin   = (float*)(w + o); o += rnd256(sizeof(float) * (size_t)N * 256);
    float* agg2    = (float*)(w + o); o += rnd256(sizeof(float) * (size_t)N * 256);
    // h1 is reused as post-LN activations (h1b); h2lin reused as post-LN h2;
    // agg2 reused for tanh projection t.

    float* out_drug = (float*)d_out;
    float* out_food = out_drug + (size_t)B * 256;
    float* out_attn = out_food + (size_t)B * 256;

    const int T = 256;
    #define GRID(n) dim3(((n) + T - 1) / T)

    // ---- self-loop mean edge-attr + CSR build --------------------------------
    k_fill_i<<<GRID(N), T, 0, stream>>>(degI, 0, N);
    k_fill_f<<<GRID(N * 6), T, 0, stream>>>(eamean, 0.0f, N * 6);
    k_deg_ea<<<GRID(E), T, 0, stream>>>(ei, ea, degI, eamean, E);
    k_mean_ea<<<GRID(N * 6), T, 0, stream>>>(eamean, degI, N);
    k_scan<<<1, 1024, 0, stream>>>(degI, offsets, N);
    k_fill_i<<<GRID(N), T, 0, stream>>>(cursor, 0, N);
    k_csr_fill<<<GRID(EF), T, 0, stream>>>(ei, offsets, cursor, csr_e, csr_s, E, EF);

    // ---- layer 1: GAT (heads=4, C=256, concat) -------------------------------
    k_weae<<<6 * H1, 256, 0, stream>>>(We1, ae1, weae1, H1, 1024);
    k_gemm_wmma_f32<<<dim3(N / 64, 1024 / 64), 256, 0, stream>>>(x, W1, h1, N, 55, 1024, 0);
    k_rowdot<<<GRID(N * H1 * 32), T, 0, stream>>>(h1, as1, hs, N, H1);
    k_rowdot<<<GRID(N * H1 * 32), T, 0, stream>>>(h1, ad1, hd, N, H1);
    k_fill_f<<<GRID(N * H1), T, 0, stream>>>(segmax, -FLT_MAX, N * H1);
    k_fill_f<<<GRID(N * H1), T, 0, stream>>>(segsum, 0.0f, N * H1);
    k_logits<<<GRID(EF * H1), T, 0, stream>>>(ei, ea, eamean, hs, hd, weae1, att, segmax, E, N, H1);
    k_expsum<<<GRID(EF * H1), T, 0, stream>>>(ei, att, segmax, segsum, E, N, H1);
    k_aggregate<<<N, 256, 0, stream>>>(offsets, csr_e, csr_s, att, segsum, h1, agg1, H1);
    k_bias_ln<<<N, 256, 0, stream>>>(agg1, b1, g1, be1, h1, 1024, 1);   // relu, h1 := h1b

    // ---- layer 2: GAT (heads=1, C=256, mean==identity) -----------------------
    k_gemm_wmma_f32<<<dim3(N / 64, 256 / 64), 256, 0, stream>>>(h1, W2, h2lin, N, 1024, 256, 0);
    k_weae<<<6, 256, 0, stream>>>(We2, ae2, weae2, 1, 256);
    k_rowdot<<<GRID(N * 32), T, 0, stream>>>(h2lin, as2, hs, N, 1);
    k_rowdot<<<GRID(N * 32), T, 0, stream>>>(h2lin, ad2, hd, N, 1);
    k_fill_f<<<GRID(N), T, 0, stream>>>(segmax, -FLT_MAX, N);
    k_fill_f<<<GRID(N), T, 0, stream>>>(segsum, 0.0f, N);
    k_logits<<<GRID(EF), T, 0, stream>>>(ei, ea, eamean, hs, hd, weae2, att, segmax, E, N, 1);
    k_expsum<<<GRID(EF), T, 0, stream>>>(ei, att, segmax, segsum, E, N, 1);
    k_aggregate<<<N, 256, 0, stream>>>(offsets, csr_e, csr_s, att, segsum, h2lin, agg2, 1);
    k_bias_ln<<<N, 256, 0, stream>>>(agg2, b2, g2, be2, h2lin, 256, 0);  // h2lin := h2

    // ---- attention pooling ---------------------------------------------------
    k_gemm_wmma_f32<<<dim3(N / 64, 256 / 64), 256, 0, stream>>>(h2lin, Wa, agg2, N, 256, 256, 1); // tanh
    k_rowdot<<<GRID(N * 32), T, 0, stream>>>(agg2, va, scores, N, 1);
    k_pool<<<B, 256, 0, stream>>>(h2lin, scores, out_drug, out_attn);
    k_food<<<GRID(B * 256), T, 0, stream>>>(fids, emb, out_food, B);

    (void)n_in; (void)out_size; (void)ws_size;
    #undef GRID
}